// InferenceLayer_56642028699714
// MI455X (gfx1250) — compile-verified
//
#include <hip/hip_runtime.h>

typedef __attribute__((ext_vector_type(2))) float v2f;
typedef __attribute__((ext_vector_type(8))) float v8f;

#define K_CLS 21
#define D_IN  768
#define T_LEN 512
#define B_SZ  64

// ---------------------------------------------------------------------------
// Kernel 1: logits[M][21] = vectors[M][768] @ W[21][768]^T + b, M = B*T = 32768
// fp32 WMMA (V_WMMA_F32_16X16X4_F32): exact fp32, memory-bound at 23.3 TB/s.
// Each block: 4 waves, each wave owns one 16-row M tile, two 16-col N tiles.
// B operand staged in LDS pre-swizzled into the native WMMA B layout.
// ---------------------------------------------------------------------------
__global__ __launch_bounds__(128) void proj_wmma_kernel(
    const float* __restrict__ A, const float* __restrict__ W,
    const float* __restrict__ bias, float* __restrict__ logits)
{
  // Bt[d2][n] = { W[n][2*d2], W[n][2*d2+1] }, n padded to 32 (zeros for n>=21)
  __shared__ v2f Bt[(D_IN / 2) * 32];   // 384*32 float2 = 96 KB

  const int tid = threadIdx.x;
  for (int idx = tid; idx < (D_IN / 2) * 32; idx += 128) {
    const int d2 = idx >> 5;
    const int n  = idx & 31;
    v2f v = {0.f, 0.f};
    if (n < K_CLS) {
      v.x = W[n * D_IN + 2 * d2];
      v.y = W[n * D_IN + 2 * d2 + 1];
    }
    Bt[idx] = v;
  }
  __syncthreads();

  const int wave    = tid >> 5;
  const int lane    = tid & 31;
  const int l15     = lane & 15;
  const int hi      = lane >> 4;           // half-wave selects K pair {0,1} vs {2,3}
  const int rowbase = blockIdx.x * 64 + wave * 16;

  // A-operand layout: lane(l15,hi) holds A[row l15][kb + 2*hi .. +1] (float2)
  const float* aptr = A + (size_t)(rowbase + l15) * D_IN + 2 * hi;

  v8f acc0 = {};  // cols 0..15
  v8f acc1 = {};  // cols 16..31 (only 16..20 valid)

  for (int kbo = 0; kbo < D_IN; kbo += 32) {
    __builtin_prefetch(aptr + kbo + 256, 0, 1);   // global_prefetch_b8, ~1KB ahead
#pragma unroll
    for (int ku = 0; ku < 32; ku += 4) {
      const int kb = kbo + ku;
      v2f a  = *(const v2f*)(aptr + kb);
      const int d2 = (kb >> 1) + hi;
      v2f b0 = Bt[d2 * 32 + l15];
      v2f b1 = Bt[d2 * 32 + 16 + l15];
      acc0 = __builtin_amdgcn_wmma_f32_16x16x4_f32(false, a, false, b0,
                                                   (short)0, acc0, false, false);
      acc1 = __builtin_amdgcn_wmma_f32_16x16x4_f32(false, a, false, b1,
                                                   (short)0, acc1, false, false);
    }
  }

  // C/D layout: VGPR r, lanes 0-15 -> M=r, lanes 16-31 -> M=r+8; N = l15
#pragma unroll
  for (int r = 0; r < 8; ++r) {
    const int row = rowbase + r + 8 * hi;
    logits[(size_t)row * K_CLS + l15] = acc0[r] + bias[l15];
    const int c1 = 16 + l15;
    if (c1 < K_CLS) logits[(size_t)row * K_CLS + c1] = acc1[r] + bias[c1];
  }
}

// ---------------------------------------------------------------------------
// Kernel 2: CRF log-likelihood. One wave per batch element. Lane j owns
// alpha[j]; transition column trans[:,j] lives in registers; alpha broadcast
// via wave32 shuffles. Gold-path score lane-parallel over t.
// ---------------------------------------------------------------------------
__global__ __launch_bounds__(32) void crf_kernel(
    const float* __restrict__ logits, const int* __restrict__ tags,
    const int* __restrict__ mask, const float* __restrict__ trans,
    const float* __restrict__ start_t, const float* __restrict__ end_t,
    float* __restrict__ ws)
{
  const int b     = blockIdx.x;
  const int j     = threadIdx.x;
  const bool valid = (j < K_CLS);
  const int base  = b * T_LEN;
  const float NEG = -1e30f;

  float tcol[K_CLS];
#pragma unroll
  for (int i = 0; i < K_CLS; ++i) tcol[i] = valid ? trans[i * K_CLS + j] : 0.f;

  float alpha = valid ? (start_t[j] + logits[(size_t)base * K_CLS + j]) : NEG;

  for (int t = 1; t < T_LEN; ++t) {
    const float* lt = logits + (size_t)(base + t) * K_CLS;
    const float emit = valid ? lt[j] : 0.f;
    const float m    = (float)mask[base + t];

    float av[K_CLS];
    float mx = NEG;
#pragma unroll
    for (int i = 0; i < K_CLS; ++i) {
      av[i] = __shfl(alpha, i, 32) + tcol[i];
      mx = fmaxf(mx, av[i]);
    }
    float s = 0.f;
#pragma unroll
    for (int i = 0; i < K_CLS; ++i) s += __expf(av[i] - mx);
    const float na = mx + __logf(s) + emit;
    if (valid && m > 0.5f) alpha = na;   // alpha = na*m + alpha*(1-m), m in {0,1}
  }

  // log-denominator: logsumexp(alpha + end_trans) across lanes 0..20
  float v  = valid ? (alpha + end_t[j]) : NEG;
  float mx = v;
  for (int off = 16; off; off >>= 1) mx = fmaxf(mx, __shfl_xor(mx, off, 32));
  float e = valid ? __expf(v - mx) : 0.f;
  for (int off = 16; off; off >>= 1) e += __shfl_xor(e, off, 32);
  const float denom = mx + __logf(e);

  // gold-path score, lane-parallel over t
  float spar = 0.f, msum = 0.f;
  for (int t = j; t < T_LEN; t += 32) {
    const int   tg = tags[base + t];
    const float m  = (float)mask[base + t];
    msum += m;
    if (t < T_LEN - 1) {
      const int   tgn = tags[base + t + 1];
      const float mn  = (float)mask[base + t + 1];
      spar += trans[tg * K_CLS + tgn] * mn
            + logits[(size_t)(base + t) * K_CLS + tg] * m;
    }
  }
  for (int off = 16; off; off >>= 1) {
    spar += __shfl_xor(spar, off, 32);
    msum += __shfl_xor(msum, off, 32);
  }

  if (j == 0) {
    int last_idx = (int)msum - 1;
    if (last_idx < 0) last_idx = 0;
    const int last_tag = tags[base + last_idx];
    const float mlast  = (float)mask[base + T_LEN - 1];
    const float score  = start_t[tags[base]] + spar + end_t[last_tag]
                       + logits[(size_t)(base + T_LEN - 1) * K_CLS + last_tag] * mlast;
    atomicAdd(&ws[0], score - denom);
    atomicAdd(&ws[1], msum);
  }
}

__global__ void init_ws_kernel(float* ws) { ws[0] = 0.f; ws[1] = 0.f; }

__global__ void finalize_kernel(const float* __restrict__ ws, float* __restrict__ out)
{
  out[0] = -ws[0] / ws[1];
}

// ---------------------------------------------------------------------------
extern "C" void kernel_launch(void* const* d_in, const int* in_sizes, int n_in,
                              void* d_out, int out_size, void* d_ws, size_t ws_size,
                              hipStream_t stream)
{
  const float* vectors = (const float*)d_in[0];   // [64,512,768] f32
  const int*   targets = (const int*)  d_in[1];   // [64,512] i32
  const int*   maskp   = (const int*)  d_in[2];   // [64,512] i32
  const float* W       = (const float*)d_in[3];   // [21,768] f32
  const float* bias    = (const float*)d_in[4];   // [21] f32
  const float* trans   = (const float*)d_in[5];   // [21,21] f32
  const float* st      = (const float*)d_in[6];   // [21] f32
  const float* et      = (const float*)d_in[7];   // [21] f32

  float* out    = (float*)d_out;   // [0] = loss, [1..] = logits [64,512,21]
  float* logits = out + 1;
  float* ws     = (float*)d_ws;    // [0] = sum ll, [1] = sum mask

  init_ws_kernel<<<1, 1, 0, stream>>>(ws);
  proj_wmma_kernel<<<(B_SZ * T_LEN) / 64, 128, 0, stream>>>(vectors, W, bias, logits);
  crf_kernel<<<B_SZ, 32, 0, stream>>>(logits, targets, maskp, trans, st, et, ws);
  finalize_kernel<<<1, 1, 0, stream>>>(ws, out);
}